// HardQuantizationThresholdRoundingLayer_9088150798505
// MI455X (gfx1250) — compile-verified
//
#include <hip/hip_runtime.h>
#include <stdint.h>

// Problem constants from the reference: B=4096, F=512, T=63
#define B_DIM   4096
#define F_DIM   512
#define T_DIM   63
#define F_TILE  256     // features per block (== blockDim.x)
#define ROWS    32      // batch rows per block
#define NTHREADS 256    // 8 wave32 waves per block

#if defined(__has_builtin)
#  if __has_builtin(__builtin_amdgcn_global_load_async_to_lds_b32)
#    define USE_ASYNC_LDS 1
#  endif
#  if __has_builtin(__builtin_amdgcn_global_load_async_to_lds_b128)
#    define USE_ASYNC_LDS_B128 1
#  endif
#  if __has_builtin(__builtin_amdgcn_s_wait_asynccnt)
#    define HAVE_WAIT_ASYNC_BUILTIN 1
#  endif
#endif

// Builtin signatures (confirmed by hipcc diagnostics):
//   b32 : (AS1 int*  src, AS3 int*  dst, imm offset, imm cpol)
//   b128: (AS1 int4* src, AS3 int4* dst, imm offset, imm cpol)
typedef __attribute__((address_space(1))) int* as1_iptr;
typedef __attribute__((address_space(3))) int* as3_iptr;
typedef __attribute__((ext_vector_type(4))) int v4i;
typedef __attribute__((address_space(1))) v4i* as1_v4ptr;
typedef __attribute__((address_space(3))) v4i* as3_v4ptr;

__global__ __launch_bounds__(NTHREADS)
void quant_round_kernel(const float* __restrict__ x,
                        const float* __restrict__ thr,
                        float* __restrict__ out)
{
    // Threshold tile for this block's 256 features: 256*63*4 = 63 KB LDS.
    // 16-byte aligned so b128 async copies are legal.
    __shared__ __align__(16) float sTh[F_TILE * T_DIM];

    const int tid = threadIdx.x;
    const int f0  = blockIdx.x * F_TILE;
    const float* gth = thr + (size_t)f0 * T_DIM;   // contiguous 16128 floats,
                                                   // byte offset 0 or 64512 (16B-aligned)

#if USE_ASYNC_LDS_B128
    // CDNA5 async global->LDS copy, 16 B per lane per op (ASYNCcnt-tracked).
    // 16128 floats = 4032 float4 -> ~16 ops per thread.
    for (int i = tid; i < (F_TILE * T_DIM) / 4; i += NTHREADS) {
        __builtin_amdgcn_global_load_async_to_lds_b128(
            (as1_v4ptr)(uintptr_t)(gth + 4 * i),
            (as3_v4ptr)(sTh + 4 * i),
            /*offset=*/0, /*cpol=*/0);
    }
#  if HAVE_WAIT_ASYNC_BUILTIN
    __builtin_amdgcn_s_wait_asynccnt(0);
#  else
    asm volatile("s_wait_asynccnt 0" ::: "memory");
#  endif
#elif USE_ASYNC_LDS
    for (int i = tid; i < F_TILE * T_DIM; i += NTHREADS) {
        __builtin_amdgcn_global_load_async_to_lds_b32(
            (as1_iptr)(uintptr_t)(gth + i),
            (as3_iptr)(sTh + i),
            /*offset=*/0, /*cpol=*/0);
    }
#  if HAVE_WAIT_ASYNC_BUILTIN
    __builtin_amdgcn_s_wait_asynccnt(0);
#  else
    asm volatile("s_wait_asynccnt 0" ::: "memory");
#  endif
#else
    for (int i = tid; i < F_TILE * T_DIM; i += NTHREADS) sTh[i] = gth[i];
#endif
    __syncthreads();

    const int   base = tid * T_DIM;           // this thread's threshold row in LDS
    const int   f    = f0 + tid;              // feature handled by this thread
    const int   r0   = blockIdx.y * ROWS;
    const float* xp  = x   + (size_t)r0 * F_DIM + f;
    float*       op  = out + (size_t)r0 * F_DIM + f;

#pragma unroll 4
    for (int r = 0; r < ROWS; ++r) {
        if (r + 8 < ROWS)  // pull future rows toward the WGP (global_prefetch_b8)
            __builtin_prefetch(xp + (size_t)(r + 8) * F_DIM, 0, 0);

        // Streaming, one-touch data: non-temporal hint.
        float xv = __builtin_nontemporal_load(xp + (size_t)r * F_DIM);

        // Branchless binary search over 63 sorted thresholds:
        // j = #{ t : xv > th[t] }  in [0, 63]
        int j = (xv > sTh[base + 31]) ? 32 : 0;
        if (xv > sTh[base + j + 15]) j += 16;
        if (xv > sTh[base + j + 7])  j += 8;
        if (xv > sTh[base + j + 3])  j += 4;
        if (xv > sTh[base + j + 1])  j += 2;
        if (xv > sTh[base + j])      j += 1;

        // rounded[j]:
        //   j==0   : th0   + (th0 - th1 )*0.5 = th0 - (th1-th0)*0.5
        //   1..62  : th[j-1] + (th[j]-th[j-1])*0.5
        //   j==63  : th62  + (th62 - th61)*0.5
        int p = j < 1 ? 1 : (j > 62 ? 62 : j);
        float lo = sTh[base + p - 1];
        float hi = sTh[base + p];
        float s      = (j == 0)  ? -0.5f : 0.5f;
        float anchor = (j == 63) ? hi    : lo;
        // mul by +/-0.5 is exact -> fma is bit-identical to the reference's
        // separate (d*0.5) then add.
        float res = fmaf(hi - lo, s, anchor);

        __builtin_nontemporal_store(res, op + (size_t)r * F_DIM);
    }
}

extern "C" void kernel_launch(void* const* d_in, const int* in_sizes, int n_in,
                              void* d_out, int out_size, void* d_ws, size_t ws_size,
                              hipStream_t stream)
{
    (void)in_sizes; (void)n_in; (void)out_size; (void)d_ws; (void)ws_size;
    const float* x   = (const float*)d_in[0];   // (B, F) f32
    const float* thr = (const float*)d_in[1];   // (F, T) f32, sorted along T
    float* out = (float*)d_out;                 // (B, F) f32

    dim3 grid(F_DIM / F_TILE, B_DIM / ROWS);    // (2, 128) = 256 blocks
    quant_round_kernel<<<grid, NTHREADS, 0, stream>>>(x, thr, out);
}